// TransformerDecoderKVCache_38603166056862
// MI455X (gfx1250) — compile-verified
//
#include <hip/hip_runtime.h>
#include <stdint.h>

// ---------------------------------------------------------------------------
// KV-cache append: out = concat([k_prev, k_new], dim=1) ++ concat([v_prev, v_new], dim=1)
// B=2, S_PREV=8192, S_NEW=16, H=32, D=128, f32.
// Zero FLOPs, ~1.05 GB of traffic -> HBM-bound, ~45us floor at 23.3 TB/s.
// Strategy: 128-bit non-temporal streaming copy (working set 5.5x the 192MB L2),
// uniform per-block segment selection, and the CDNA5 async global<->LDS DMA
// path (ASYNCcnt) for the small "new" tiles.
// ---------------------------------------------------------------------------

typedef float v4f __attribute__((ext_vector_type(4)));

constexpr long PREV = 8192L * 32 * 128;       // 33,554,432 floats per (tensor,batch) prev block
constexpr long NEWN = 16L * 32 * 128;         //     65,536 floats per (tensor,batch) new block
constexpr long CAT  = PREV + NEWN;            // floats per batch in concatenated output
constexpr long HALF = 2 * CAT;                // floats in k_cat (== floats in v_cat)
constexpr long SEG4 = PREV / 4;               // float4s per prev segment = 2^23

// ---------------------------------------------------------------------------
// Bulk copy of the 4 big prev segments (2 x 128MB k, 2 x 128MB v).
// gridDim.y = 4 selects the segment -> k/v pointer pick is a scalar cselect,
// loop body is a pure 4x-unrolled NT B128 copy with immediate offsets
// (0/4K/8K/12K bytes, within the s24 instruction offset).
// Launch: dim3(2048, 4) x 256 threads. Per segment: 2^23 float4s,
// stride chunk = 2048 blocks * 1024 float4 = 2^21 -> exactly 4 iterations.
// ---------------------------------------------------------------------------
__global__ __launch_bounds__(256) void kv_append_prev(
    const float* __restrict__ k_prev,
    const float* __restrict__ v_prev,
    float* __restrict__ out)
{
    const int seg   = blockIdx.y;        // 0..3 : (k,b0) (k,b1) (v,b0) (v,b1)
    const int which = seg >> 1;          // 0=k, 1=v   (uniform -> SGPR select)
    const int b     = seg & 1;

    const v4f* __restrict__ s4 =
        (const v4f*)((which ? v_prev : k_prev) + (long)b * PREV);
    v4f* __restrict__ d4 =
        (v4f*)(out + (long)which * HALF + (long)b * CAT);

    long base         = (long)blockIdx.x * 1024 + threadIdx.x;
    const long stride = (long)gridDim.x * 1024;   // float4s per stride step

    for (; base < SEG4; base += stride) {
        v4f a0 = __builtin_nontemporal_load(s4 + base);
        v4f a1 = __builtin_nontemporal_load(s4 + base + 256);
        v4f a2 = __builtin_nontemporal_load(s4 + base + 512);
        v4f a3 = __builtin_nontemporal_load(s4 + base + 768);
        __builtin_nontemporal_store(a0, d4 + base);
        __builtin_nontemporal_store(a1, d4 + base + 256);
        __builtin_nontemporal_store(a2, d4 + base + 512);
        __builtin_nontemporal_store(a3, d4 + base + 768);
    }
}

// ---------------------------------------------------------------------------
// The 4 small "new" segments (2 x 256KB k_new, 2 x 256KB v_new) go through the
// CDNA5 async global<->LDS DMA path: GLOBAL_LOAD_ASYNC_TO_LDS_B128 /
// GLOBAL_STORE_ASYNC_FROM_LDS_B128, tracked by ASYNCcnt and fenced with
// s_wait_asynccnt. Each thread owns one 16B LDS slot per segment.
// 64 blocks x 256 threads = 16384 threads = NEWN/4 float4s per segment.
// ---------------------------------------------------------------------------
__global__ __launch_bounds__(256) void kv_append_new_async(
    const float* __restrict__ k_new,
    const float* __restrict__ v_new,
    float* __restrict__ out)
{
    __shared__ char stage[256 * 16 * 4];     // 16 KB: 4 segments x 256 threads x 16B

    const unsigned t = blockIdx.x * 256u + threadIdx.x;   // 0 .. 16383

    // LDS byte address of this thread's slot 0 (low 32 bits of the generic
    // pointer to LDS are the workgroup-relative LDS address per the ISA
    // aperture rule). Also keeps `stage` live.
    const unsigned lbase = (unsigned)(uintptr_t)(&stage[threadIdx.x * 16]);

    // Issue 4 async HBM -> LDS loads (ASYNCcnt += 4)
    #pragma unroll
    for (int seg = 0; seg < 4; ++seg) {
        const float* srcp = (seg & 2) ? v_new : k_new;                     // uniform SGPR pair
        const unsigned src_off = ((unsigned)(seg & 1) * (unsigned)NEWN + t * 4u) * 4u; // bytes
        const unsigned lds_off = lbase + (unsigned)seg * (256u * 16u);
        asm volatile("global_load_async_to_lds_b128 %0, %1, %2"
                     :
                     : "v"(lds_off), "v"(src_off), "s"(srcp)
                     : "memory");
    }

    // Wait for all async loads to land in LDS
    asm volatile("s_wait_asynccnt 0x0" ::: "memory");

    // Issue 4 async LDS -> HBM stores
    #pragma unroll
    for (int seg = 0; seg < 4; ++seg) {
        const int which = seg >> 1;  // 0=k, 1=v
        const int b     = seg & 1;
        const unsigned long long dst_elem =
            (unsigned long long)which * (unsigned long long)HALF +
            (unsigned long long)b     * (unsigned long long)CAT  +
            (unsigned long long)PREV  + (unsigned long long)t * 4ull;
        const unsigned dst_off = (unsigned)(dst_elem * 4ull);              // bytes (<538MB, fits u32)
        const unsigned lds_off = lbase + (unsigned)seg * (256u * 16u);
        asm volatile("global_store_async_from_lds_b128 %0, %1, %2"
                     :
                     : "v"(dst_off), "v"(lds_off), "s"(out)
                     : "memory");
    }

    // Drain async stores before wave exit
    asm volatile("s_wait_asynccnt 0x0" ::: "memory");
}

// ---------------------------------------------------------------------------
extern "C" void kernel_launch(void* const* d_in, const int* in_sizes, int n_in,
                              void* d_out, int out_size, void* d_ws, size_t ws_size,
                              hipStream_t stream)
{
    const float* k_prev = (const float*)d_in[0];
    const float* v_prev = (const float*)d_in[1];
    const float* k_new  = (const float*)d_in[2];
    const float* v_new  = (const float*)d_in[3];
    float* out = (float*)d_out;

    // 512 MB bulk: 2048 blocks/segment x 4 segments, 256 threads,
    // 4x-unrolled NT B128 copy, exactly 4 loop iterations per thread.
    kv_append_prev<<<dim3(2048, 4, 1), 256, 0, stream>>>(k_prev, v_prev, out);

    // 1 MB of new tokens through the async global<->LDS DMA path
    kv_append_new_async<<<64, 256, 0, stream>>>(k_new, v_new, out);
}